// ConvTransduce1D_46102178955737
// MI455X (gfx1250) — compile-verified
//
#include <hip/hip_runtime.h>

// Problem constants (fixed by the reference).
#define BB   16
#define TT   4096
#define CC   128
#define WW   4096      // output window count (stride 1, pad 2, K=5)
#define KN   52        // 26 one-char + 26 two-char tokens
#define TW   128       // windows per block
#define NTH  256       // threads per block: 2 halves x TW windows
#define PITCH 28       // LDS floats per row (channels 0..27; stride 28 -> 2-way conflict max)
#define ROWS (TW + 4)  // halo of 2 rows each side
#define NEGV (-1e30f)

typedef unsigned int u32;
typedef unsigned long long u64;
typedef __attribute__((ext_vector_type(4))) u32 u32x4_t;
typedef __attribute__((ext_vector_type(8))) int i32x8_t;
typedef __attribute__((ext_vector_type(4))) int i32x4_t;

#if __has_builtin(__builtin_amdgcn_tensor_load_to_lds) && defined(__HIP_DEVICE_COMPILE__)
#define USE_TDM 1
#else
#define USE_TDM 0
#endif

// 2-way LSE: max + log1p(exp(-|a-b|))  -> 1 exp + 1 log (vs 2 exp + 1 log)
__device__ __forceinline__ float lse2(float a, float b) {
  float m = fmaxf(a, b);
  float d = fabsf(a - b);
  return m + __logf(1.0f + __expf(-d));
}
__device__ __forceinline__ float lse3(float a, float b, float c) {
  float m = fmaxf(fmaxf(a, b), c);
  return m + __logf(__expf(a - m) + __expf(b - m) + __expf(c - m));
}

__global__ __launch_bounds__(NTH) void conv_transduce_kernel(
    const float* __restrict__ x, float* __restrict__ out) {
  __shared__ __align__(16) float smem[ROWS * PITCH];     // 14784 B emission tile
  __shared__ __align__(16) float out_lds[TW * KN];       // 26624 B output tile

  const int tid = threadIdx.x;
  const int tilesPerB = WW / TW;                // 32
  const int b  = blockIdx.x / tilesPerB;
  const int w0 = (blockIdx.x % tilesPerB) * TW;

  const int r0      = w0 - 2;                  // first (possibly negative) global row
  const int r_start = r0 < 0 ? 0 : r0;
  const int top_pad = r_start - r0;            // 0..2 rows to zero at LDS top

#if USE_TDM
  // ---- Tensor Data Mover: DMA tile x[b, r_start : ..., 0:28] -> LDS ----
  // Zero the top halo rows (only first tile of each b needs it).
  for (int i = tid; i < top_pad * PITCH; i += NTH) smem[i] = 0.0f;

  if (tid < 32) {  // one wave issues the TDM op (EXEC is ignored by tensor ops)
    const u32 lds_base =
        (u32)(u64)(void*)smem + (u32)(top_pad * PITCH * 4);
    const u64 gaddr =
        (u64)(const void*)(x + ((size_t)b * TT + r_start) * CC);
    const int nrows      = ROWS - top_pad;     // rows the TDM writes
    const int rows_avail = TT - r_start;       // tensor_dim1: rows past end read as 0

    u32x4_t g0;
    g0[0] = 1u;                                            // count=1, user mode
    g0[1] = lds_base;                                      // lds_addr [63:32]
    g0[2] = (u32)(gaddr & 0xFFFFFFFFu);                    // global_addr lo
    g0[3] = (u32)((gaddr >> 32) & 0x01FFFFFFu)             // global_addr hi (57b)
            | (2u << 30);                                  // type=2 ("image")

    i32x8_t g1;
    g1[0] = (int)(2u << 16);                               // data_size=2 -> 4 bytes
    g1[1] = (int)(((u32)CC & 0xFFFFu) << 16);              // tensor_dim0[15:0]=128
    g1[2] = (int)(((u32)rows_avail & 0xFFFFu) << 16);      // dim0 hi=0 | tensor_dim1 lo
    g1[3] = (int)((((u32)rows_avail >> 16) & 0xFFFFu)      // tensor_dim1 hi
            | ((u32)PITCH << 16));                         // tile_dim0 = 28
    g1[4] = (int)((u32)nrows & 0xFFFFu);                   // tile_dim1 | tile_dim2=0
    g1[5] = (int)CC;                                       // tensor_dim0_stride = 128
    g1[6] = 0;                                             // stride hi | dim1_stride lo
    g1[7] = 0;

    i32x4_t g2 = {0, 0, 0, 0};                             // 2D tensor: unused groups
    i32x4_t g3 = {0, 0, 0, 0};
    i32x8_t g4 = {0, 0, 0, 0, 0, 0, 0, 0};                 // extra group (6-arg form)

    __builtin_amdgcn_tensor_load_to_lds(g0, g1, g2, g3, g4, 0);
    __builtin_amdgcn_s_wait_tensorcnt(0);
  }
  __syncthreads();
#else
  // ---- Fallback: cooperative staged load (also used for host-side parse) ----
  for (int i = tid; i < ROWS * PITCH; i += NTH) {
    int rl = i / PITCH, ch = i % PITCH;
    int g = r0 + rl;
    smem[i] = (g >= 0 && g < TT) ? x[((size_t)b * TT + g) * CC + ch] : 0.0f;
  }
  __syncthreads();
#endif

  // ---- Per-thread: window w = w0 + w_local; half 0 -> c=1..13, half 1 -> c=14..26
  const int w_local = tid & (TW - 1);
  const int half    = tid >> 7;
  const int c_begin = 1 + half * 13;

  float emB[5];
#pragma unroll
  for (int t = 0; t < 5; ++t) emB[t] = smem[(w_local + t) * PITCH + 0];

  float* __restrict__ orow = out_lds + w_local * KN;

  float cur[5], nxt[5];
#pragma unroll
  for (int t = 0; t < 5; ++t) cur[t] = smem[(w_local + t) * PITCH + c_begin];

  for (int c = c_begin; c < c_begin + 13; ++c) {
    // channel c+1 (second char of the 2-char token; reused as `cur` next iter)
#pragma unroll
    for (int t = 0; t < 5; ++t) nxt[t] = smem[(w_local + t) * PITCH + (c + 1)];

    // --- 1-char token [c]: states 0,1,2; arcs (0,0,B)(0,1,c)(1,1,c)(1,2,B)(2,2,B)
    {
      float a0 = 0.0f, a1 = NEGV, a2 = NEGV;
#pragma unroll
      for (int t = 0; t < 5; ++t) {
        float n1 = cur[t] + lse2(a0, a1);
        float n2 = emB[t] + lse2(a1, a2);
        a0 += emB[t];
        a1 = n1;
        a2 = n2;
      }
      orow[c - 1] = lse2(a1, a2);  // accept states 1,2
    }

    // --- 2-char token [c,c+1]: states 0..4
    //     extra arcs: (2,3,c+1)(3,3,c+1)(1,3,c+1 skip)(3,4,B)(4,4,B)
    {
      float b0 = 0.0f, b1 = NEGV, b2 = NEGV, b3 = NEGV, b4 = NEGV;
#pragma unroll
      for (int t = 0; t < 5; ++t) {
        float n1 = cur[t] + lse2(b0, b1);
        float n2 = emB[t] + lse2(b1, b2);
        float n3 = nxt[t] + lse3(b1, b2, b3);
        float n4 = emB[t] + lse2(b3, b4);
        b0 += emB[t];
        b1 = n1; b2 = n2; b3 = n3; b4 = n4;
      }
      orow[26 + (c - 1)] = lse2(b3, b4);  // accept states 3,4
    }

#pragma unroll
    for (int t = 0; t < 5; ++t) cur[t] = nxt[t];
  }

  __syncthreads();

  // ---- Coalesced, vectorized writeback of the contiguous (TW x KN) tile ----
  {
    float4* __restrict__ dst =
        (float4*)(out + ((size_t)b * WW + w0) * KN);
    const float4* __restrict__ src = (const float4*)out_lds;
#pragma unroll 2
    for (int i = tid; i < (TW * KN) / 4; i += NTH) dst[i] = src[i];
  }
}

extern "C" void kernel_launch(void* const* d_in, const int* in_sizes, int n_in,
                              void* d_out, int out_size, void* d_ws, size_t ws_size,
                              hipStream_t stream) {
  (void)in_sizes; (void)n_in; (void)d_ws; (void)ws_size; (void)out_size;
  const float* x = (const float*)d_in[0];
  float* out = (float*)d_out;
  dim3 grid(BB * (WW / TW));   // 16 * 32 = 512 blocks
  dim3 block(NTH);             // 256 threads = 8 wave32
  conv_transduce_kernel<<<grid, block, 0, stream>>>(x, out);
}